// BatchTrainableButterfly_60395830116593
// MI455X (gfx1250) — compile-verified
//
#include <hip/hip_runtime.h>
#include <hip/hip_bf16.h>
#include <math.h>

// ---------------------------------------------------------------------------
// Trainable butterfly (L=1024) over 4096 tokens x 4 mesh batches.
// Transform is linear => build dense 1024x1024 complex operator per mesh
// batch, stored TRANSPOSED as WT[b][n][k] (32MB, L2-resident), then
// OUT[b] = X (4096x1024) * W_b as 4 complex fp32 GEMMs on
// v_wmma_f32_16x16x4_f32. HBM-bound (~160MB => ~7us at 23.3 TB/s).
// Transposed storage makes BOTH A and B fragments contiguous float2 LDS
// loads; 32Mx32N per-wave tiles give ~2 WMMA per DS op in the K-loop.
// ---------------------------------------------------------------------------

typedef __attribute__((ext_vector_type(2))) float v2f;
typedef __attribute__((ext_vector_type(8))) float v8f;

#define L1024 1024
#define NTOK  4096
#define NB    4

// ---- LDS address helper: generic -> raw LDS byte offset (AS3 ptrtoint) ----
typedef __attribute__((address_space(3))) float lds_f;
__device__ __forceinline__ unsigned lds_off(float* p) {
  return (unsigned)(unsigned long long)(lds_f*)p;
}

// ---- CDNA5 async global->LDS staging (tracked by ASYNCcnt) ----
__device__ __forceinline__ void async_b128(unsigned ldsaddr, const float* g) {
  unsigned long long ga = (unsigned long long)g;
  asm volatile("global_load_async_to_lds_b128 %0, %1, off"
               :: "v"(ldsaddr), "v"(ga) : "memory");
}
__device__ __forceinline__ void wait_async0() {
  asm volatile("s_wait_asynccnt 0x0" ::: "memory");
}

__device__ __forceinline__ unsigned brev10(unsigned x) { return __brev(x) >> 22; }

// inverse of the reference's forward_indices permutation:
// old position o -> new position, for level < 9 (bs = 4<<level)
__device__ __forceinline__ int fwd_scatter(int o, int level) {
  int bs   = 4 << level;
  int half = bs >> 1;
  int blk  = o & ~(bs - 1);
  int w    = o - blk;
  int c    = (w >= half) ? 1 : 0;
  int r    = w & (half - 1);
  return blk + 2 * r + c;
}

// ===========================================================================
// Kernel 1: build WT[b][n][k] = T_b(e_k)[n].  grid=(1024 k, 4 b), block=256.
// Each block runs the full butterfly pipeline on one basis vector in LDS.
// Transposed writes are scattered per-thread but W is L2-resident (32MB).
// ===========================================================================
__global__ void __launch_bounds__(256)
butterfly_build_W(const float* __restrict__ phases,
                  float* __restrict__ WTre, float* __restrict__ WTim) {
  const int k   = blockIdx.x;
  const int b   = blockIdx.y;
  const int tid = threadIdx.x;

  __shared__ float2 buf[L1024];

  // init: x = e_k bit-reversed  => single 1 at index br(k)
  for (int e = tid; e < L1024; e += 256) buf[e] = make_float2(0.f, 0.f);
  __syncthreads();
  if (tid == 0) buf[brev10((unsigned)k)] = make_float2(1.f, 0.f);
  __syncthreads();

  const float* phB = phases + (size_t)b * 11 * 512 * 2;  // phases[b]

  for (int level = 0; level < 10; ++level) {
    const float* phL = phB + level * 1024;  // [j][c] flat
    float2 y[2][2];
    int js[2] = {tid, tid + 256};
    #pragma unroll
    for (int p = 0; p < 2; ++p) {
      int j = js[p];
      float2 a0 = buf[2 * j];
      float2 a1 = buf[2 * j + 1];
      float s0, c0, s1, c1;
      __sincosf(phL[2 * j + 0], &s0, &c0);
      __sincosf(phL[2 * j + 1], &s1, &c1);
      // a' = a * (c + i s)
      float2 A0 = make_float2(a0.x * c0 - a0.y * s0, a0.x * s0 + a0.y * c0);
      float2 A1 = make_float2(a1.x * c1 - a1.y * s1, a1.x * s1 + a1.y * c1);
      // y0 = A0 + i*A1 ; y1 = i*A0 + A1
      y[p][0] = make_float2(A0.x - A1.y, A0.y + A1.x);
      y[p][1] = make_float2(A1.x - A0.y, A1.y + A0.x);
    }
    __syncthreads();
    #pragma unroll
    for (int p = 0; p < 2; ++p) {
      int o0 = 2 * js[p], o1 = 2 * js[p] + 1;
      int t0 = (level < 9) ? fwd_scatter(o0, level) : o0;
      int t1 = (level < 9) ? fwd_scatter(o1, level) : o1;
      buf[t0] = y[p][0];
      buf[t1] = y[p][1];
    }
    __syncthreads();
  }

  // final: multiply by e^{i ph[9] flat}, bit-reverse, /sqrt(1024)
  const float* phF = phB + 9 * 1024;
  const float inv = 0.03125f;  // 1/32
  #pragma unroll
  for (int q = 0; q < 4; ++q) {
    int n = tid + q * 256;
    int e = (int)brev10((unsigned)n);  // out[n] = buf_ph[br(n)]
    float s, c;
    __sincosf(phF[e], &s, &c);
    float2 a = buf[e];
    float re = (a.x * c - a.y * s) * inv;
    float im = (a.x * s + a.y * c) * inv;
    size_t idx = ((size_t)b * L1024 + n) * L1024 + k;  // TRANSPOSED: [b][n][k]
    WTre[idx] = re;
    WTim[idx] = im;
  }
}

// ===========================================================================
// Kernel 2: complex GEMM  OUT[b] = X * W_b   via v_wmma_f32_16x16x4_f32.
// Block tile 64(M) x 128(N), BK=32, 256 threads = 8 waves (2 M x 4 N),
// each wave owns a 32x32 complex tile: 2x2 subtiles of 16x16 (re+im accum).
// Both A and B fragments are contiguous float2 LDS loads ([row][k] layout).
// grid = (N/128=8, M/64=64, b=4)
// ===========================================================================
__global__ void __launch_bounds__(256)
butterfly_gemm(const float* __restrict__ Xre, const float* __restrict__ Xim,
               const float* __restrict__ WTre, const float* __restrict__ WTim,
               float2* __restrict__ out) {
  const int nBase = blockIdx.x * 128;
  const int mBase = blockIdx.y * 64;
  const int b     = blockIdx.z;
  const int tid   = threadIdx.x;
  const int lane  = tid & 31;
  const int wave  = tid >> 5;
  const int mw    = (wave & 1) * 32;   // wave's M offset in tile
  const int nw    = (wave >> 1) * 32;  // wave's N offset in tile
  const int half  = lane >> 4;
  const int l15   = lane & 15;

  __shared__ __align__(16) float sXre[64 * 32];    // [m][k]  8KB
  __shared__ __align__(16) float sXim[64 * 32];    //         8KB
  __shared__ __align__(16) float sWre[128 * 32];   // [n][k] 16KB
  __shared__ __align__(16) float sWim[128 * 32];   //        16KB

  v8f acc_re[2][2], acc_im[2][2];
  #pragma unroll
  for (int mt = 0; mt < 2; ++mt)
    #pragma unroll
    for (int nt = 0; nt < 2; ++nt) {
      acc_re[mt][nt] = (v8f)0.f;
      acc_im[mt][nt] = (v8f)0.f;
    }

  const float* WreB = WTre + (size_t)b * L1024 * L1024;
  const float* WimB = WTim + (size_t)b * L1024 * L1024;

  for (int k0 = 0; k0 < L1024; k0 += 32) {
    // ---- async stage: X tile 64x32 [m][k], WT tile 128x32 [n][k] ----
    #pragma unroll
    for (int q = 0; q < 2; ++q) {    // X: 512 x 16B segments per array
      int e = tid + q * 256;
      int row = e >> 3, seg = (e & 7) * 4;
      const float* gr = Xre + (size_t)(mBase + row) * L1024 + k0 + seg;
      const float* gi = Xim + (size_t)(mBase + row) * L1024 + k0 + seg;
      async_b128(lds_off(&sXre[row * 32 + seg]), gr);
      async_b128(lds_off(&sXim[row * 32 + seg]), gi);
    }
    #pragma unroll
    for (int q = 0; q < 4; ++q) {    // WT: 1024 x 16B segments per array
      int e = tid + q * 256;
      int row = e >> 3, seg = (e & 7) * 4;
      const float* gr = WreB + (size_t)(nBase + row) * L1024 + k0 + seg;
      const float* gi = WimB + (size_t)(nBase + row) * L1024 + k0 + seg;
      async_b128(lds_off(&sWre[row * 32 + seg]), gr);
      async_b128(lds_off(&sWim[row * 32 + seg]), gi);
    }
    wait_async0();
    __syncthreads();

    // ---- compute: 8 WMMA K-steps of 4, 16 wmma per step per wave ----
    #pragma unroll
    for (int kk = 0; kk < 32; kk += 4) {
      const int ka = kk + 2 * half;  // frag: lane<16 -> K0,K1 ; lane>=16 -> K2,K3
      v2f ar[2], ai[2], ain[2], br_[2], bi_[2];
      #pragma unroll
      for (int mt = 0; mt < 2; ++mt) {
        ar[mt] = *(const v2f*)&sXre[(mw + mt * 16 + l15) * 32 + ka];
        ai[mt] = *(const v2f*)&sXim[(mw + mt * 16 + l15) * 32 + ka];
        ain[mt].x = -ai[mt].x;       // for  -Aim*Bim  (f32 WMMA has no A-neg)
        ain[mt].y = -ai[mt].y;
      }
      #pragma unroll
      for (int nt = 0; nt < 2; ++nt) {
        br_[nt] = *(const v2f*)&sWre[(nw + nt * 16 + l15) * 32 + ka];
        bi_[nt] = *(const v2f*)&sWim[(nw + nt * 16 + l15) * 32 + ka];
      }
      #pragma unroll
      for (int mt = 0; mt < 2; ++mt)
        #pragma unroll
        for (int nt = 0; nt < 2; ++nt) {
          // OUTre += Are*Bre - Aim*Bim ; OUTim += Are*Bim + Aim*Bre
          acc_re[mt][nt] = __builtin_amdgcn_wmma_f32_16x16x4_f32(
              false, ar[mt],  false, br_[nt], (short)0, acc_re[mt][nt], false, false);
          acc_re[mt][nt] = __builtin_amdgcn_wmma_f32_16x16x4_f32(
              false, ain[mt], false, bi_[nt], (short)0, acc_re[mt][nt], false, false);
          acc_im[mt][nt] = __builtin_amdgcn_wmma_f32_16x16x4_f32(
              false, ar[mt],  false, bi_[nt], (short)0, acc_im[mt][nt], false, false);
          acc_im[mt][nt] = __builtin_amdgcn_wmma_f32_16x16x4_f32(
              false, ai[mt],  false, br_[nt], (short)0, acc_im[mt][nt], false, false);
        }
    }
    __syncthreads();
  }

  // ---- epilogue: C/D layout -> interleaved complex64 out[b][m][n] ----
  #pragma unroll
  for (int mt = 0; mt < 2; ++mt)
    #pragma unroll
    for (int nt = 0; nt < 2; ++nt) {
      int n = nBase + nw + nt * 16 + l15;
      #pragma unroll
      for (int r = 0; r < 8; ++r) {
        int m = mBase + mw + mt * 16 + half * 8 + r;
        out[((size_t)b * NTOK + m) * L1024 + n] =
            make_float2(acc_re[mt][nt][r], acc_im[mt][nt][r]);
      }
    }
}

// ===========================================================================
extern "C" void kernel_launch(void* const* d_in, const int* in_sizes, int n_in,
                              void* d_out, int out_size, void* d_ws, size_t ws_size,
                              hipStream_t stream) {
  const float* x_re   = (const float*)d_in[0];  // 4096 x 1024 f32
  const float* x_im   = (const float*)d_in[1];  // 4096 x 1024 f32
  const float* phases = (const float*)d_in[2];  // 4 x 11 x 512 x 2 f32

  float* WTre = (float*)d_ws;                              // 4*1024*1024 f32
  float* WTim = WTre + (size_t)NB * L1024 * L1024;         // 4*1024*1024 f32

  // 1) build dense per-batch operators, transposed (32MB, L2-resident)
  butterfly_build_W<<<dim3(L1024, NB), 256, 0, stream>>>(phases, WTre, WTim);

  // 2) 4 complex GEMMs: OUT[b] = X * W_b   (M=4096, N=1024, K=1024)
  butterfly_gemm<<<dim3(L1024 / 128, NTOK / 64, NB), 256, 0, stream>>>(
      x_re, x_im, WTre, WTim, (float2*)d_out);
}